// EPLoss_77610059039196
// MI455X (gfx1250) — compile-verified
//
#include <hip/hip_runtime.h>
#include <math.h>

// Problem constants (match reference setup_inputs).
#define NB 192     // batch
#define NY 48      // n_y
#define NT 48      // n_T
#define NC 6625    // C
#define SLAB ((size_t)NT * NY * NB)   // elements per coefficient array

typedef __attribute__((ext_vector_type(2))) float v2f;
typedef __attribute__((ext_vector_type(8))) float v8f;

// ---------------------------------------------------------------------------
// Kernel A: parallel gather + coefficient build.
//   p_C[t][j][b] = R[b,j,0] * max(pred[b,j,target[b,t]], 0)
//   p_I[t][j][b] = (j==NY-1 ? 1 : R[b,j,1]) * max(I[b,j,target[b,t]], 0)
//   p_D[t][j][b] = (target[b,t]==1) ? 1 : R[b,j,2]
// [t][j][b] layout => kernel B reads are coalesced across lanes (b-contig).
// ---------------------------------------------------------------------------
__global__ void ep_gather(const float* __restrict__ pred,
                          const float* __restrict__ R,
                          const float* __restrict__ I,
                          const int*   __restrict__ target,
                          float* __restrict__ pC,
                          float* __restrict__ pI,
                          float* __restrict__ pD) {
  const int b = blockIdx.x;
  for (int idx = threadIdx.x; idx < NT * NY; idx += blockDim.x) {
    const int t = idx / NY;
    const int j = idx - t * NY;
    const int tgt = target[b * NT + t];
    const size_t src = (size_t)(b * NY + j) * NC + (size_t)tgt;
    const float pg = fmaxf(pred[src], 0.0f);
    const float ig = fmaxf(I[src], 0.0f);
    const int rb = (b * NY + j) * 3;
    const float r0 = R[rb + 0];
    const float r1 = R[rb + 1];
    const float r2 = R[rb + 2];
    const size_t o = ((size_t)t * NY + j) * NB + b;
    pC[o] = r0 * pg;
    pI[o] = (j == NY - 1 ? 1.0f : r1) * ig;
    pD[o] = (tgt == 1) ? 1.0f : r2;
  }
}

// ---------------------------------------------------------------------------
// Kernel B: one lane per batch b (192 threads = 6 wave32).
// row[48] lives in VGPRs; 47 outer steps, serial inner linear recurrence.
// Final mean(log) reduced by wave 0 via 3x V_WMMA_F32_16X16X4_F32 with an
// all-ones B matrix (rows of D = row-sums of A), then one shfl_xor(16).
// ---------------------------------------------------------------------------
__global__ void __launch_bounds__(NB)
ep_dp(const float* __restrict__ pC,
      const float* __restrict__ pI,
      const float* __restrict__ pD,
      float* __restrict__ out) {
  __shared__ float logs[NB];
  const int b = threadIdx.x;

  float row[NY];
  // row0 = cumprod([1, pD[0,1], ..., pD[0,NY-1]])
  row[0] = 1.0f;
#pragma unroll
  for (int j = 1; j < NY; ++j)
    row[j] = row[j - 1] * pD[((size_t)0 * NY + j) * NB + b];

  for (int t = 1; t < NT; ++t) {
    const float* __restrict__ pIs = pI + (size_t)(t - 1) * NY * NB + b;
    const float* __restrict__ pCs = pC + (size_t)(t - 1) * NY * NB + b;
    const float* __restrict__ pDs = pD + (size_t)t * NY * NB + b;
    if (t + 1 < NT) {
      // global_prefetch_b8: pull next t-slice toward the WGP while the
      // dependent FMA chain of this step drains.
      __builtin_prefetch(pI + (size_t)t * NY * NB + b, 0, 0);
      __builtin_prefetch(pC + (size_t)t * NY * NB + b, 0, 0);
      __builtin_prefetch(pD + (size_t)(t + 1) * NY * NB + b, 0, 0);
    }
    float prev_old = row[0];            // row_{t-1}[j-1]
    float nr = prev_old * pIs[0];       // new_row[0] = a[0]
    row[0] = nr;
#pragma unroll
    for (int j = 1; j < NY; ++j) {
      const float old_j = row[j];
      // a[j] = row[j]*pI[t-1,j] + row[j-1]*pC[t-1,j-1]
      const float a = fmaf(old_j, pIs[(size_t)j * NB],
                           prev_old * pCs[(size_t)(j - 1) * NB]);
      // new_row[j] = a[j] + new_row[j-1]*pD[t,j-1]
      nr = fmaf(nr, pDs[(size_t)(j - 1) * NB], a);
      row[j] = nr;
      prev_old = old_j;
    }
  }

  logs[b] = logf(row[NY - 1]);
  __syncthreads();

  // Wave 0 (lanes 0..31, EXEC all-1s) reduces 192 values with the matrix unit:
  // three chained f32 WMMAs against an all-ones B; D[i][n] accumulates the
  // row-sums, identical across n. Fold lane halves with shfl_xor(16).
  if (threadIdx.x < 32) {
    const int L = threadIdx.x;
    const int m = L & 15;            // M row of the A tile
    const int kh = (L >> 4) << 1;    // K pair: lanes 0-15 -> K{0,1}, 16-31 -> K{2,3}
    v8f acc = {};
    const v2f ones = {1.0f, 1.0f};
#pragma unroll
    for (int T = 0; T < 3; ++T) {
      v2f a;
      a.x = logs[T * 64 + m * 4 + kh + 0];
      a.y = logs[T * 64 + m * 4 + kh + 1];
      acc = __builtin_amdgcn_wmma_f32_16x16x4_f32(
          /*neg_a=*/false, a, /*neg_b=*/false, ones,
          /*c_mod=*/(short)0, acc, /*reuse_a=*/false, /*reuse_b=*/false);
    }
    // acc[v], lanes 0-15: D[M=v][*]; lanes 16-31: D[M=v+8][*]
    float s = acc[0] + acc[1] + acc[2] + acc[3] + acc[4] + acc[5] + acc[6] + acc[7];
    s += __shfl_xor(s, 16, 32);
    if (L == 0) out[0] = s * (1.0f / (float)NB);
  }
}

// ---------------------------------------------------------------------------
extern "C" void kernel_launch(void* const* d_in, const int* in_sizes, int n_in,
                              void* d_out, int out_size, void* d_ws, size_t ws_size,
                              hipStream_t stream) {
  (void)in_sizes; (void)n_in; (void)out_size; (void)ws_size;
  const float* pred   = (const float*)d_in[0];
  const float* R      = (const float*)d_in[1];
  const float* I      = (const float*)d_in[2];
  const int*   target = (const int*)d_in[3];

  float* pC = (float*)d_ws;       // 3 * NT*NY*NB * 4B = ~5.3 MB scratch
  float* pI = pC + SLAB;
  float* pD = pI + SLAB;

  ep_gather<<<NB, 256, 0, stream>>>(pred, R, I, target, pC, pI, pD);
  ep_dp<<<1, NB, 0, stream>>>(pC, pI, pD, (float*)d_out);
}